// CrossEntropyLoss_32066225832638
// MI455X (gfx1250) — compile-verified
//
#include <hip/hip_runtime.h>
#include <cmath>

typedef float v2f __attribute__((ext_vector_type(2)));
typedef float v8f __attribute__((ext_vector_type(8)));

#define EMB_D 128
#define WAVES_PER_BLOCK 8
#define NNEG 4

__device__ __forceinline__ float softplus_f(float x) {
    // numerically stable: max(x,0) + log1p(exp(-|x|))
    return fmaxf(x, 0.0f) + log1pf(expf(-fabsf(x)));
}

__device__ __forceinline__ float pick8(v8f c, int j) {
    float r = c[0];
    if (j == 1) r = c[1];
    if (j == 2) r = c[2];
    if (j == 3) r = c[3];
    if (j == 4) r = c[4];
    if (j == 5) r = c[5];
    if (j == 6) r = c[6];
    if (j == 7) r = c[7];
    return r;
}

// One wave computes 16 edge dot-products <h[sidx[base+m]], h[didx[base+m]]>
// via V_WMMA_F32_16X16X4_F32 accumulation over K=128, writing scores to out[0..15].
__device__ __forceinline__ void tile_scores_wmma(
    const float* __restrict__ h, const int* __restrict__ sidx,
    const int* __restrict__ didx, long base, int lane,
    float* __restrict__ out /* 16 floats in LDS */) {
    const int r    = lane & 15;        // row (A) / column (B) this lane serves
    const int koff = (lane >> 4) << 1; // K sub-offset: lanes 16-31 handle K+2,K+3
    const int is = sidx[base + r];
    const int id = didx[base + r];
    const float* ps = h + (long)is * EMB_D + koff;
    const float* pd = h + (long)id * EMB_D + koff;
    v8f c = {0.f, 0.f, 0.f, 0.f, 0.f, 0.f, 0.f, 0.f};
#pragma unroll
    for (int k = 0; k < EMB_D; k += 4) {
        v2f a = *(const v2f*)(ps + k);  // A: 16x4 f32 fragment (2 VGPRs)
        v2f b = *(const v2f*)(pd + k);  // B: 4x16 f32 fragment (2 VGPRs)
        // (neg_a, A, neg_b, B, c_mod, C, reuse_a, reuse_b)
        c = __builtin_amdgcn_wmma_f32_16x16x4_f32(false, a, false, b,
                                                  (short)0, c, false, false);
    }
    // C layout: VGPR j -> lanes 0-15: (M=j, N=lane); lanes 16-31: (M=j+8, N=lane-16).
    // Diagonal m: lane m / VGPR m for m<8; lane m+16 / VGPR m-8 for m>=8.
    if (lane < 8) {
        out[lane] = pick8(c, lane);
    } else if (lane >= 24) {
        out[lane - 16] = pick8(c, lane - 24);
    }
}

__global__ void edge_bce_mrr_kernel(const float* __restrict__ h,
                                    const int* __restrict__ pos_src,
                                    const int* __restrict__ pos_dst,
                                    const int* __restrict__ neg_src,
                                    const int* __restrict__ neg_dst,
                                    float* __restrict__ acc, int nPosTiles) {
    __shared__ float buf[WAVES_PER_BLOCK][16 + 16 * NNEG];
    __shared__ float accL, accM;
    const int tid  = threadIdx.x;
    const int lane = tid & 31;
    const int w    = tid >> 5;
    if (tid == 0) { accL = 0.f; accM = 0.f; }

    const long T = (long)blockIdx.x * WAVES_PER_BLOCK + w;  // pos-tile index
    const bool active = (T < (long)nPosTiles);              // wave-uniform
    if (active) {
        // 16 positive edges of this tile
        tile_scores_wmma(h, pos_src, pos_dst, T * 16, lane, &buf[w][0]);
        // their 64 negatives (contiguous: pos edge e owns negs 4e..4e+3)
        for (int t = 0; t < NNEG; ++t) {
            tile_scores_wmma(h, neg_src, neg_dst, T * 64 + (long)t * 16, lane,
                             &buf[w][16 + t * 16]);
        }
    }
    __syncthreads();

    float lossv = 0.f, mrrv = 0.f;
    if (active && lane < 16) {
        const float p = buf[w][lane];
        lossv = softplus_f(p) - p;  // label = 1
        int gt = 0;
#pragma unroll
        for (int j = 0; j < NNEG; ++j) {
            const float n = buf[w][16 + NNEG * lane + j];
            lossv += softplus_f(n);  // label = 0
            gt += (n > p) ? 1 : 0;   // stable argsort: pos wins ties
        }
        mrrv = 1.0f / (float)(1 + gt);
    }
    // wave32 reduction
#pragma unroll
    for (int off = 16; off > 0; off >>= 1) {
        lossv += __shfl_xor(lossv, off, 32);
        mrrv  += __shfl_xor(mrrv, off, 32);
    }
    if (lane == 0) {
        atomicAdd(&accL, lossv);
        atomicAdd(&accM, mrrv);
    }
    __syncthreads();
    if (tid == 0) {
        atomicAdd(&acc[0], accL);
        atomicAdd(&acc[1], accM);
    }
}

__global__ void init_acc_kernel(float* __restrict__ acc) {
    if (threadIdx.x < 2) acc[threadIdx.x] = 0.f;
}

__global__ void finalize_kernel(const float* __restrict__ acc,
                                float* __restrict__ out, float invTotal,
                                float invPos) {
    out[0] = acc[0] * invTotal;  // loss = sum / (E_pos + E_neg)
    out[1] = acc[1] * invPos;    // mrr  = sum / E_pos
}

extern "C" void kernel_launch(void* const* d_in, const int* in_sizes, int n_in,
                              void* d_out, int out_size, void* d_ws,
                              size_t ws_size, hipStream_t stream) {
    const float* h      = (const float*)d_in[0];
    const int* pos_src  = (const int*)d_in[1];
    const int* pos_dst  = (const int*)d_in[2];
    const int* neg_src  = (const int*)d_in[3];
    const int* neg_dst  = (const int*)d_in[4];
    const int nPos = in_sizes[1];
    const int nNeg = in_sizes[3];

    float* acc = (float*)d_ws;
    float* out = (float*)d_out;

    const int tiles  = nPos / 16;  // E_POS = 262144 -> 16384 tiles
    const int blocks = (tiles + WAVES_PER_BLOCK - 1) / WAVES_PER_BLOCK;

    init_acc_kernel<<<1, 32, 0, stream>>>(acc);
    edge_bce_mrr_kernel<<<blocks, 32 * WAVES_PER_BLOCK, 0, stream>>>(
        h, pos_src, pos_dst, neg_src, neg_dst, acc, tiles);
    finalize_kernel<<<1, 1, 0, stream>>>(acc, out,
                                         1.0f / (float)(nPos + nNeg),
                                         1.0f / (float)nPos);
}